// Splatting_19258633355983
// MI455X (gfx1250) — compile-verified
//
#include <hip/hip_runtime.h>

// Splatting (forward warp scatter-add) for B=8, C=3, H=W=1024, fp32.
// Strategy: per-32x32-tile LDS privatization with a 6px halo.
//  - in-tile splats  -> ds_add_f32 (LDS float atomics, 64 banks/WGP)
//  - LDS flush       -> coalesced global_atomic_add_f32 (output is L2-resident: 100MB < 192MB L2)
//  - far splats (|flow|>6, P~6e-7) -> direct global atomic fallback (correctness for any flow)

#define C_CH   3
#define H_SZ   1024
#define W_SZ   1024
#define TILE   32
#define HALO   6
#define RW     (TILE + 2 * HALO)          // 44
#define RCELLS (RW * RW)                  // 1936
#define LDS_N  (C_CH * RCELLS)            // 5808 floats = 23232 B
#define NT     256

__global__ void splat_zero_kernel(float4* __restrict__ out, int n4) {
    int i = blockIdx.x * blockDim.x + threadIdx.x;
    if (i < n4) out[i] = make_float4(0.f, 0.f, 0.f, 0.f);
}

__global__ __launch_bounds__(NT) void splat_tile_kernel(const float* __restrict__ frame,
                                                        const float* __restrict__ flow,
                                                        float* __restrict__ out) {
    __shared__ float tile[LDS_N];

    const int tid         = threadIdx.x;
    const int tilesX      = W_SZ / TILE;
    const int tilesPerImg = tilesX * (H_SZ / TILE);
    const int b           = blockIdx.x / tilesPerImg;
    const int t           = blockIdx.x % tilesPerImg;
    const int tileY0      = (t / tilesX) * TILE;
    const int tileX0      = (t % tilesX) * TILE;

    for (int r = tid; r < LDS_N; r += NT) tile[r] = 0.f;
    __syncthreads();

    const long HW    = (long)H_SZ * W_SZ;
    const float* fU  = flow  + (long)b * 2 * HW;        // flow[b,0]
    const float* fV  = fU + HW;                         // flow[b,1]
    const float* fr  = frame + (long)b * C_CH * HW;     // frame[b,0]
    float* outB      = out   + (long)b * C_CH * HW;     // out[b,0]

    #pragma unroll
    for (int it = 0; it < (TILE * TILE) / NT; ++it) {
        int p  = it * NT + tid;
        int px = p % TILE;
        int py = p / TILE;
        int gx = tileX0 + px;
        int gy = tileY0 + py;
        long idx = (long)gy * W_SZ + gx;

        float u  = fU[idx];
        float v  = fV[idx];
        float f0 = fr[idx];
        float f1 = fr[idx + HW];
        float f2 = fr[idx + 2 * HW];

        float fx  = (float)gx + u;
        float fy  = (float)gy + v;
        float x0f = floorf(fx);
        float y0f = floorf(fy);
        int   x0  = (int)x0f;
        int   y0  = (int)y0f;
        float dx  = fx - x0f;
        float dy  = fy - y0f;
        float wx[2] = {1.f - dx, dx};
        float wy[2] = {1.f - dy, dy};

        #pragma unroll
        for (int cy = 0; cy < 2; ++cy) {
            #pragma unroll
            for (int cx = 0; cx < 2; ++cx) {
                int   xi = x0 + cx;
                int   yi = y0 + cy;
                float w  = wx[cx] * wy[cy];
                int   lx = xi - tileX0 + HALO;
                int   ly = yi - tileY0 + HALO;
                if ((unsigned)lx < (unsigned)RW && (unsigned)ly < (unsigned)RW) {
                    // Common path: lands in the LDS tile (+halo). No image-bounds
                    // check needed here: out-of-image halo cells are dropped at flush.
                    int base = ly * RW + lx;
                    atomicAdd(&tile[base],              f0 * w);   // ds_add_f32
                    atomicAdd(&tile[base + RCELLS],     f1 * w);
                    atomicAdd(&tile[base + 2 * RCELLS], f2 * w);
                } else if ((unsigned)xi < (unsigned)W_SZ && (unsigned)yi < (unsigned)H_SZ) {
                    // Rare far-splat fallback (|flow| > HALO).
                    long o = (long)yi * W_SZ + xi;
                    unsafeAtomicAdd(outB + o,          f0 * w);   // global_atomic_add_f32
                    unsafeAtomicAdd(outB + o + HW,     f1 * w);
                    unsafeAtomicAdd(outB + o + 2 * HW, f2 * w);
                }
            }
        }
    }
    __syncthreads();

    // Flush LDS accumulator to global with coalesced atomics; skip zeros
    // (outer halo cells are mostly untouched) and out-of-image halo cells.
    for (int r = tid; r < LDS_N; r += NT) {
        float val = tile[r];
        if (val != 0.f) {
            int c   = r / RCELLS;
            int rem = r - c * RCELLS;
            int ry  = rem / RW;
            int rx  = rem - ry * RW;
            int gx  = tileX0 - HALO + rx;
            int gy  = tileY0 - HALO + ry;
            if ((unsigned)gx < (unsigned)W_SZ && (unsigned)gy < (unsigned)H_SZ) {
                long o = ((long)c * H_SZ + gy) * W_SZ + gx;
                unsafeAtomicAdd(outB + o, val);
            }
        }
    }
}

extern "C" void kernel_launch(void* const* d_in, const int* in_sizes, int n_in,
                              void* d_out, int out_size, void* d_ws, size_t ws_size,
                              hipStream_t stream) {
    const float* frame = (const float*)d_in[0];   // [B,3,1024,1024]
    const float* flow  = (const float*)d_in[1];   // [B,2,1024,1024]
    float* out         = (float*)d_out;           // [B,3,1024,1024]

    const int B = in_sizes[0] / (C_CH * H_SZ * W_SZ);

    // 1) Zero the output (harness poisons d_out with 0xAA).
    int n4 = out_size / 4;
    splat_zero_kernel<<<(n4 + NT - 1) / NT, NT, 0, stream>>>((float4*)out, n4);

    // 2) Tiled scatter-add.
    const int tilesPerImg = (W_SZ / TILE) * (H_SZ / TILE);   // 1024
    splat_tile_kernel<<<B * tilesPerImg, NT, 0, stream>>>(frame, flow, out);
}